// SOM_59742995087529
// MI455X (gfx1250) — compile-verified
//
#include <hip/hip_runtime.h>
#include <stdint.h>

// SOM step: M=N=256 grid, DIM=512, ALPHA=0.3, SIGMA=128, NUM_EPOCHS=100.
#define SOM_DIM 512
#define SOM_N   256

typedef __attribute__((ext_vector_type(2))) float v2f;
typedef __attribute__((ext_vector_type(8))) float v8f;

// ---------------------------------------------------------------------------
// Kernel 0: reset the global argmin key (d_ws is poisoned by the harness and
// must be re-initialized every call for determinism).
// ---------------------------------------------------------------------------
__global__ void som_init(unsigned long long* ws) { *ws = ~0ULL; }

// ---------------------------------------------------------------------------
// Kernel 1: BMU search. One wave32 owns 16 weight rows. For each K-chunk of 4,
// A[m,k] = w[m,k] - x[k]; D += A * A^T via V_WMMA_F32_16X16X4_F32. Because the
// CDNA5 A(16x4) and B(4x16) VGPR layouts are transposes of each other, B = A^T
// reuses the identical register pair -> one float2 load per lane per chunk,
// each weight byte read exactly once. diag(D) = squared distances.
// ---------------------------------------------------------------------------
__global__ __launch_bounds__(256) void som_bmu(const float* __restrict__ w,
                                               const float* __restrict__ x,
                                               unsigned long long* __restrict__ ws) {
    __shared__ float xs[SOM_DIM];
    __shared__ unsigned long long blockmin;
    if (threadIdx.x == 0) blockmin = ~0ULL;
    for (int i = threadIdx.x; i < SOM_DIM; i += 256) xs[i] = x[i];
    __syncthreads();

    const int lane = threadIdx.x & 31;
    const int wave = threadIdx.x >> 5;
    const int base = (blockIdx.x * 8 + wave) * 16;   // 16 rows per wave
    const int m    = lane & 15;                      // row within tile
    const int half = lane >> 4;                      // 0: K=0,1  1: K=2,3
    const float* wrow = w + (size_t)(base + m) * SOM_DIM + half * 2;

    v8f acc = {};
    #pragma unroll 4
    for (int k0 = 0; k0 < SOM_DIM; k0 += 4) {
        const float2 wv = *(const float2*)(wrow + k0);   // 8B aligned
        const int c = k0 + half * 2;
        v2f a;
        a.x = wv.x - xs[c];
        a.y = wv.y - xs[c + 1];
        // D = A * A^T + C ; A and B share the same registers (B = A^T layout).
        acc = __builtin_amdgcn_wmma_f32_16x16x4_f32(
            /*neg_a=*/false, a, /*neg_b=*/false, a,
            /*c_mod=*/(short)0, acc, /*reuse_a=*/false, /*reuse_b=*/false);
    }

    // Diagonal extraction (C layout: VGPR v = rows v and v+8):
    //   m in 0..7  -> acc[m]   at lane m        (lanes 0..7)
    //   m in 8..15 -> acc[m-8] at lane m+8      (lanes 16..23)
    const int j = lane & 7;
    float d2 = acc[0];
    #pragma unroll
    for (int t = 1; t < 8; ++t) if (j == t) d2 = acc[t];

    const bool valid = (lane < 8) || (lane >= 16 && lane < 24);
    const int row = base + ((lane < 16) ? j : (8 + j));
    // Non-negative f32 bits are order-isomorphic; u64-min => first-index ties,
    // matching jnp.argmin. Invalid lanes contribute identity.
    const unsigned long long key = valid
        ? (((unsigned long long)__float_as_uint(d2) << 32) | (unsigned int)row)
        : ~0ULL;

    atomicMin(&blockmin, key);        // ds_min_u64 within the block
    __syncthreads();
    if (threadIdx.x == 0) atomicMin(ws, blockmin);   // 512 global atomics total
}

// ---------------------------------------------------------------------------
// Kernel 2: neighbourhood update + emit bmu_loc. Pure streaming: one float2
// per thread. Weights region of d_out starts at +2 floats (8B aligned), so
// 8-byte vectors keep every store aligned.
// ---------------------------------------------------------------------------
__global__ __launch_bounds__(256) void som_update(const float* __restrict__ w,
                                                  const float* __restrict__ x,
                                                  const int* __restrict__ epoch_p,
                                                  const unsigned long long* __restrict__ ws,
                                                  float* __restrict__ out) {
    const unsigned long long key = *ws;
    const unsigned int bmu = (unsigned int)(key & 0xFFFFFFFFu);
    const int br = (int)(bmu >> 8);
    const int bc = (int)(bmu & 255u);

    const float lrate    = 1.0f - (float)(*epoch_p) * 0.01f;   // 1 - epoch/100
    const float alpha_op = 0.3f * lrate;
    const float sigma_op = 128.0f * lrate;
    const float inv_s2   = 1.0f / (sigma_op * sigma_op);

    const int idx2 = blockIdx.x * 256 + threadIdx.x;   // float2 index
    const int row  = idx2 >> 8;                        // 256 float2 per row
    const int col2 = idx2 & 255;
    const int r = row >> 8;
    const int c = row & 255;
    const int dd = (r - br) * (r - br) + (c - bc) * (c - bc);
    const float lm = alpha_op * expf(-(float)dd * inv_s2);

    const float2 wv = ((const float2*)w)[idx2];
    const float2 xv = ((const float2*)x)[col2];
    float2 o;
    o.x = wv.x + lm * (xv.x - wv.x);
    o.y = wv.y + lm * (xv.y - wv.y);
    ((float2*)(out + 2))[idx2] = o;

    if (idx2 == 0) { out[0] = (float)br; out[1] = (float)bc; }
}

// ---------------------------------------------------------------------------
extern "C" void kernel_launch(void* const* d_in, const int* in_sizes, int n_in,
                              void* d_out, int out_size, void* d_ws, size_t ws_size,
                              hipStream_t stream) {
    const float* x      = (const float*)d_in[0];   // input_vector [512]
    const float* w      = (const float*)d_in[1];   // weights [65536*512]
    const int*   epoch  = (const int*)d_in[2];     // scalar epoch
    float* out          = (float*)d_out;           // [2 + 65536*512]
    unsigned long long* ws = (unsigned long long*)d_ws;

    som_init<<<1, 1, 0, stream>>>(ws);
    // 65536 rows / (8 waves * 16 rows) = 512 blocks of 256 threads
    som_bmu<<<512, 256, 0, stream>>>(w, x, ws);
    // 65536 rows * 256 float2 = 16,777,216 threads
    som_update<<<65536, 256, 0, stream>>>(w, x, epoch, ws, out);
}